// Batched_STHD_SpGAT_Cosine_43301860278533
// MI455X (gfx1250) — compile-verified
//
#include <hip/hip_runtime.h>
#include <math.h>

typedef __attribute__((ext_vector_type(2))) float v2f;
typedef __attribute__((ext_vector_type(8))) float v8f;

#define N_SUB       100000
#define NUM_CLASSES 32
#define NUM_GENES   1024
#define N_EDGES     1600000
#define HID         8
#define NPART       256
#define N_TILES     (N_SUB / 16)          /* 6250 exact */
#define KQ          (NUM_GENES / 4)       /* 256 k-steps of 4 */
#define BFRAG_N     (KQ * 3 * 32)         /* 24576 float2 entries */

// ---------- order-preserving float <-> uint for atomicMax segment-max ----------
__device__ __forceinline__ unsigned f2o(float f) {
    unsigned b = __float_as_uint(f);
    return (b & 0x80000000u) ? ~b : (b | 0x80000000u);
}
__device__ __forceinline__ float o2f(unsigned u) {
    unsigned b = (u & 0x80000000u) ? (u & 0x7fffffffu) : ~u;
    return __uint_as_float(b);
}

// ---------- 0: zero accumulators / segment buffers ----------
__global__ __launch_bounds__(256) void k_init(unsigned* __restrict__ mkey,
                                              float* __restrict__ svec,
                                              float* __restrict__ llpart,
                                              float* __restrict__ cepart) {
    int i = blockIdx.x * 256 + threadIdx.x;
    if (i < N_SUB) { mkey[i] = 0u; svec[i] = 0.0f; }
    if (i < NPART) { llpart[i] = 0.0f; cepart[i] = 0.0f; }
}

// ---------- 1: ||Mu_c|| for the 32 prototypes ----------
__global__ __launch_bounds__(256) void k_munorm(const float* __restrict__ Mu,
                                                float* __restrict__ munorm) {
    __shared__ float red[256];
    int row = blockIdx.x;          // 0..31
    int t = threadIdx.x;
    const float* m = Mu + (size_t)row * NUM_GENES + t * 4;
    float s = m[0]*m[0] + m[1]*m[1] + m[2]*m[2] + m[3]*m[3];
    red[t] = s;
    __syncthreads();
    for (int k = 128; k >= 1; k >>= 1) {
        if (t < k) red[t] += red[t + k];
        __syncthreads();
    }
    if (t == 0) munorm[row] = fmaxf(sqrtf(red[0]), 1e-12f);
}

// ---------- 2: build B [1024 x 48] in WMMA fragment order ----------
// entry t = kq*96 + tile*32 + lane  ->  float2 { B[k0][c], B[k0+1][c] }
// lane<16: k0 = 4*kq,   c = 16*tile + lane
// lane>=16: k0 = 4*kq+2, c = 16*tile + lane-16   (16x4 f32 A/B VGPR layout)
__device__ __forceinline__ float bval(int k, int c,
                                      const float* Mu, const float* munorm,
                                      const float* lwl, const float* lwr) {
    if (c < 32) return Mu[(size_t)c * NUM_GENES + k] / munorm[c];
    if (c < 40) return lwl[(size_t)k * HID + (c - 32)];
    return lwr[(size_t)k * HID + (c - 40)];
}
__global__ __launch_bounds__(256) void k_bfrag(const float* __restrict__ Mu,
                                               const float* __restrict__ munorm,
                                               const float* __restrict__ lwl,
                                               const float* __restrict__ lwr,
                                               float* __restrict__ bfrag) {
    int t = blockIdx.x * 256 + threadIdx.x;
    if (t >= BFRAG_N) return;
    int lane = t & 31;
    int tile = (t >> 5) % 3;
    int kq = t / 96;
    int k0 = kq * 4 + (lane >> 4) * 2;
    int c  = tile * 16 + (lane & 15);
    v2f b;
    b.x = bval(k0,     c, Mu, munorm, lwl, lwr);
    b.y = bval(k0 + 1, c, Mu, munorm, lwl, lwr);
    ((v2f*)bfrag)[t] = b;
}

// ---------- 3: P_sub = softmax(W[subset_idx]) ; logP = log(P+1e-8) ----------
// one wave32 per row, lane = class (32 classes -> exact wave32 fit)
__global__ __launch_bounds__(256) void k_softmax(const float* __restrict__ W,
                                                 const int* __restrict__ subset,
                                                 float* __restrict__ P,
                                                 float* __restrict__ logP) {
    int wave = threadIdx.x >> 5, lane = threadIdx.x & 31;
    int i = blockIdx.x * 8 + wave;
    if (i >= N_SUB) return;
    int r = subset[i];
    float w = W[(size_t)r * NUM_CLASSES + lane];
    float mx = w;
    for (int m = 16; m >= 1; m >>= 1) mx = fmaxf(mx, __shfl_xor(mx, m, 32));
    float ex = __expf(w - mx);
    float sm = ex;
    for (int m = 16; m >= 1; m >>= 1) sm += __shfl_xor(sm, m, 32);
    float p = ex / sm;
    P[(size_t)i * NUM_CLASSES + lane] = p;
    logP[(size_t)i * NUM_CLASSES + lane] = __logf(p + 1e-8f);
}

// ---------- 4: fused f32 WMMA GEMM: Y = x_sub @ B  (48 cols) ----------
// per wave: 16-row tile, 3 accumulators (cos 0-15, cos 16-31, xl|xr),
// plus in-flight ||x_row||^2 from the same A fragments.
__global__ __launch_bounds__(256) void k_gemm(const float* __restrict__ x,
                                              const v2f* __restrict__ bf,
                                              const float* __restrict__ P,
                                              const float* __restrict__ lbl,
                                              const float* __restrict__ lbr,
                                              float* __restrict__ xl,
                                              float* __restrict__ xr,
                                              float* __restrict__ llpart) {
    int wave = threadIdx.x >> 5, lane = threadIdx.x & 31;
    int tile = blockIdx.x * 8 + wave;
    int row0 = tile * 16;
    if (row0 >= N_SUB) return;              // wave-uniform: EXEC all-1 for WMMA
    int mrow = lane & 15;
    int hi = lane >> 4;
    const float* ap = x + (size_t)(row0 + mrow) * NUM_GENES + hi * 2;

    v8f acc0 = {}; v8f acc1 = {}; v8f acc2 = {};
    float ssq = 0.0f;
    for (int kq = 0; kq < KQ; ++kq) {
        v2f a = *(const v2f*)(ap + kq * 4);
        ssq += a.x * a.x + a.y * a.y;
        v2f b0 = bf[(kq * 3 + 0) * 32 + lane];
        v2f b1 = bf[(kq * 3 + 1) * 32 + lane];
        v2f b2 = bf[(kq * 3 + 2) * 32 + lane];
        acc0 = __builtin_amdgcn_wmma_f32_16x16x4_f32(false, a, false, b0,
                                                     (short)0, acc0, false, false);
        acc1 = __builtin_amdgcn_wmma_f32_16x16x4_f32(false, a, false, b1,
                                                     (short)0, acc1, false, false);
        acc2 = __builtin_amdgcn_wmma_f32_16x16x4_f32(false, a, false, b2,
                                                     (short)0, acc2, false, false);
    }
    // lane L and L^16 split K mod 4 for row (L&15): combine, invert norm
    ssq += __shfl_xor(ssq, 16, 32);
    float inv = 1.0f / fmaxf(sqrtf(ssq), 1e-12f);   // lane L: inv-norm of row (L&15)

    float ll = 0.0f;
    int ncol = lane & 15;
    #pragma unroll
    for (int v = 0; v < 8; ++v) {
        int rowIdx = row0 + v + 8 * hi;             // C/D layout: VGPR v, hi half
        float ivv = __shfl(inv, v + 8 * hi, 32);
        const float* Pr = P + (size_t)rowIdx * NUM_CLASSES;
        float c0 = acc0[v] * ivv * 10.0f;           // cos_sim * 10, cols 0-15
        float c1 = acc1[v] * ivv * 10.0f;           // cols 16-31
        ll += c0 * Pr[ncol] + c1 * Pr[16 + ncol];
        float y = acc2[v];
        if (ncol < 8) xl[(size_t)rowIdx * HID + ncol]       = y + lbl[ncol];
        else          xr[(size_t)rowIdx * HID + (ncol - 8)] = y + lbr[ncol - 8];
    }
    for (int m = 16; m >= 1; m >>= 1) ll += __shfl_xor(ll, m, 32);
    if (lane == 0) atomicAdd(&llpart[tile & (NPART - 1)], ll);
}

// ---------- 5: per-edge attention logit + segment max ----------
__global__ __launch_bounds__(256) void k_edge1(const int* __restrict__ edges,
                                               const float* __restrict__ xl,
                                               const float* __restrict__ xr,
                                               const float* __restrict__ att,
                                               float* __restrict__ evec,
                                               unsigned* __restrict__ mkey) {
    int j = blockIdx.x * 256 + threadIdx.x;
    if (j >= N_EDGES) return;
    int src = edges[j];
    int dst = edges[N_EDGES + j];
    float4 a0 = ((const float4*)(xl + (size_t)src * HID))[0];
    float4 a1 = ((const float4*)(xl + (size_t)src * HID))[1];
    float4 b0 = ((const float4*)(xr + (size_t)dst * HID))[0];
    float4 b1 = ((const float4*)(xr + (size_t)dst * HID))[1];
    float s[8] = { a0.x + b0.x, a0.y + b0.y, a0.z + b0.z, a0.w + b0.w,
                   a1.x + b1.x, a1.y + b1.y, a1.z + b1.z, a1.w + b1.w };
    float e = 0.0f;
    #pragma unroll
    for (int h = 0; h < 8; ++h) {
        float v = s[h] > 0.0f ? s[h] : 0.2f * s[h];   // leaky_relu(0.2)
        e += att[h] * v;
    }
    evec[j] = e;
    atomicMax(&mkey[dst], f2o(e));
}

// ---------- 6: z = exp(e - m[dst]); segment sum ----------
__global__ __launch_bounds__(256) void k_edge2(const int* __restrict__ edges,
                                               const float* __restrict__ evec,
                                               const unsigned* __restrict__ mkey,
                                               float* __restrict__ zvec,
                                               float* __restrict__ svec) {
    int j = blockIdx.x * 256 + threadIdx.x;
    if (j >= N_EDGES) return;
    int dst = edges[N_EDGES + j];
    float z = __expf(evec[j] - o2f(mkey[dst]));
    zvec[j] = z;
    atomicAdd(&svec[dst], z);
}

// ---------- 7: ce += -alpha * dot(P[src], logP[dst]); one wave32 per edge ----------
__global__ __launch_bounds__(256) void k_edge3(const int* __restrict__ edges,
                                               const float* __restrict__ P,
                                               const float* __restrict__ logP,
                                               const float* __restrict__ zvec,
                                               const float* __restrict__ svec,
                                               float* __restrict__ cepart) {
    __shared__ float acc[8];
    int wave = threadIdx.x >> 5, lane = threadIdx.x & 31;
    int j = blockIdx.x * 8 + wave;
    float contrib = 0.0f;
    if (j < N_EDGES) {
        int src = edges[j];
        int dst = edges[N_EDGES + j];
        float t = P[(size_t)src * NUM_CLASSES + lane] *
                  logP[(size_t)dst * NUM_CLASSES + lane];
        for (int m = 16; m >= 1; m >>= 1) t += __shfl_xor(t, m, 32);
        float alpha = zvec[j] / svec[dst];
        contrib = -alpha * t;
    }
    if (lane == 0) acc[wave] = contrib;
    __syncthreads();
    if (threadIdx.x == 0) {
        float s2 = 0.0f;
        #pragma unroll
        for (int w = 0; w < 8; ++w) s2 += acc[w];
        atomicAdd(&cepart[blockIdx.x & (NPART - 1)], s2);
    }
}

// ---------- 8: reduce partial slots -> scalars ----------
__global__ __launch_bounds__(256) void k_final(const float* __restrict__ llpart,
                                               const float* __restrict__ cepart,
                                               float* __restrict__ out) {
    __shared__ float sl[256], sc[256];
    int t = threadIdx.x;
    sl[t] = llpart[t];
    sc[t] = cepart[t];
    __syncthreads();
    for (int m = 128; m >= 1; m >>= 1) {
        if (t < m) { sl[t] += sl[t + m]; sc[t] += sc[t + m]; }
        __syncthreads();
    }
    if (t == 0) {
        out[0] = sl[0] / (float)N_SUB;   // ll_prot
        out[1] = sc[0] / (float)N_SUB;   // ce_space
    }
}

extern "C" void kernel_launch(void* const* d_in, const int* in_sizes, int n_in,
                              void* d_out, int out_size, void* d_ws, size_t ws_size,
                              hipStream_t stream) {
    const float* x_sub  = (const float*)d_in[0];
    const float* Mu     = (const float*)d_in[1];
    const float* W      = (const float*)d_in[2];
    const float* lin_l_w = (const float*)d_in[3];
    const float* lin_l_b = (const float*)d_in[4];
    const float* lin_r_w = (const float*)d_in[5];
    const float* lin_r_b = (const float*)d_in[6];
    const float* att    = (const float*)d_in[7];
    const int* edges    = (const int*)d_in[8];
    const int* subset   = (const int*)d_in[9];

    float* out = (float*)d_out;
    float* P   = out + 2;                         // P_sub [N_SUB x 32]

    // workspace layout (floats)
    float* ws = (float*)d_ws;
    size_t off = 0;
    float* bfrag  = ws + off; off += 2 * BFRAG_N;          // 192 KB, fragment-ordered B
    float* munorm = ws + off; off += 32;
    float* xl     = ws + off; off += (size_t)N_SUB * HID;
    float* xr     = ws + off; off += (size_t)N_SUB * HID;
    float* logP   = ws + off; off += (size_t)N_SUB * NUM_CLASSES;
    float* evec   = ws + off; off += N_EDGES;
    float* zvec   = ws + off; off += N_EDGES;
    float* svec   = ws + off; off += N_SUB;
    unsigned* mkey = (unsigned*)(ws + off); off += N_SUB;
    float* llpart = ws + off; off += NPART;
    float* cepart = ws + off; off += NPART;

    k_init<<<(N_SUB + 255) / 256, 256, 0, stream>>>(mkey, svec, llpart, cepart);
    k_munorm<<<32, 256, 0, stream>>>(Mu, munorm);
    k_bfrag<<<BFRAG_N / 256, 256, 0, stream>>>(Mu, munorm, lin_l_w, lin_r_w, bfrag);
    k_softmax<<<(N_SUB + 7) / 8, 256, 0, stream>>>(W, subset, P, logP);
    k_gemm<<<(N_TILES + 7) / 8, 256, 0, stream>>>(x_sub, (const v2f*)bfrag, P,
                                                  lin_l_b, lin_r_b, xl, xr, llpart);
    k_edge1<<<N_EDGES / 256, 256, 0, stream>>>(edges, xl, xr, att, evec, mkey);
    k_edge2<<<N_EDGES / 256, 256, 0, stream>>>(edges, evec, mkey, zvec, svec);
    k_edge3<<<N_EDGES / 8, 256, 0, stream>>>(edges, P, logP, zvec, svec, cepart);
    k_final<<<1, 256, 0, stream>>>(llpart, cepart, out);
}